// LocalWLNet_83064667505070
// MI455X (gfx1250) — compile-verified
//
#include <hip/hip_runtime.h>

// ---------------------------------------------------------------------------
// LocalWLNet for MI455X (gfx1250).  fp32 end-to-end; GEMMs use
// V_WMMA_F32_16X16X4_F32 (the only fp32 WMMA shape on CDNA5).
// Graph-norm 0 is folded algebraically into W1 (affine-per-column).
// Segment sums use global float atomics (node tables are L2-resident).
// ---------------------------------------------------------------------------

typedef __attribute__((ext_vector_type(2))) float v2f;
typedef __attribute__((ext_vector_type(8))) float v8f;

#define FEAT 32   // C2

// ---------------- utility: zero a float buffer ----------------
__global__ void zero_f32_k(float* __restrict__ p, long n) {
  long i = (long)blockIdx.x * blockDim.x + threadIdx.x;
  if (i < n) p[i] = 0.0f;
}

// ---------------- per-column sum & sum-of-squares (C is power of 2) --------
__global__ void colstats_k(const float* __restrict__ X, const int* __restrict__ xidx,
                           int n, int C, float* __restrict__ stats) {
  int tid = threadIdx.x;                // blockDim.x == 256
  int col = tid & (C - 1);
  int rpb = 256 / C;                    // rows per block iteration
  int r = blockIdx.x * rpb + tid / C;
  int stride = gridDim.x * rpb;
  float s = 0.f, s2 = 0.f;
  for (; r < n; r += stride) {
    long base = (long)(xidx ? xidx[r] : r) * C;
    float v = X[base + col];
    s += v; s2 += v * v;
  }
  atomicAdd(&stats[col], s);
  atomicAdd(&stats[C + col], s2);
}

// ---------------- graph_norm -> per-column affine (scale, shift) -----------
// h = (x - a*mean) * rsqrt(var+eps) * w + b ;  var = E[x^2] - (2a - a^2)*mean^2
__global__ void gn_params_k(const float* __restrict__ stats, const float* __restrict__ a,
                            const float* __restrict__ w, const float* __restrict__ b,
                            int C, float n, float* __restrict__ scale,
                            float* __restrict__ shift) {
  int c = blockIdx.x * blockDim.x + threadIdx.x;
  if (c >= C) return;
  float mean = stats[c] / n;
  float ac = a[c];
  float var = stats[C + c] / n - (2.f * ac - ac * ac) * mean * mean;
  float sc = rsqrtf(var + 1e-5f) * w[c];
  scale[c] = sc;
  shift[c] = b[c] - ac * mean * sc;
}

// ---------------- fold gn0 affine into W1:  W1p = diag(scale)*W1, tb = shift·W1
__global__ void prep_w1_k(const float* __restrict__ W1, const float* __restrict__ scale,
                          const float* __restrict__ shift, float* __restrict__ W1p,
                          float* __restrict__ tb) {
  int t = threadIdx.x;                  // 256 threads, 1 block
  for (int i = t; i < 256 * FEAT; i += 256)
    W1p[i] = W1[i] * scale[i >> 5];
  if (t < FEAT) {
    float s = 0.f;
    for (int c = 0; c < 256; ++c) s += shift[c] * W1[c * FEAT + t];
    tb[t] = s;
  }
}

// ---------------- D(MxFEAT) = A(MxK) @ B(KxFEAT) + coladd, rows of A via xidx
// One wave per 16x16 tile; K multiple of 4; M multiple of 16; FEAT==32 (2 tiles).
__global__ void gemm_wmma_k(const float* __restrict__ A, const float* __restrict__ B,
                            float* __restrict__ D, const float* __restrict__ coladd,
                            const int* __restrict__ xidx, int M, int K) {
  int wave = (int)(((long)blockIdx.x * blockDim.x + threadIdx.x) >> 5);
  int lane = threadIdx.x & 31;
  int tm = wave >> 1;                    // FEAT/16 == 2 column tiles
  int tn = wave & 1;
  if (tm * 16 >= M) return;              // uniform per wave: EXEC stays all-ones
  int l16 = lane & 15, hi = lane >> 4;
  int row0 = tm * 16, col0 = tn * 16;
  v8f c = {0.f, 0.f, 0.f, 0.f, 0.f, 0.f, 0.f, 0.f};
#if __has_builtin(__builtin_amdgcn_wmma_f32_16x16x4_f32)
  // A 16x4 layout: lanes 0-15 hold row l16 K={k,k+1}; lanes 16-31 K={k+2,k+3}
  int arow = row0 + l16;
  long abase = (long)(xidx ? xidx[arow] : arow) * K;
  for (int kk = 0; kk < K; kk += 4) {
    int ka = kk + 2 * hi;
    v2f a, b;
    a.x = A[abase + ka];
    a.y = A[abase + ka + 1];
    b.x = B[ka * FEAT + col0 + l16];
    b.y = B[(ka + 1) * FEAT + col0 + l16];
    c = __builtin_amdgcn_wmma_f32_16x16x4_f32(false, a, false, b, (short)0, c,
                                              false, false);
  }
#else
  for (int v = 0; v < 8; ++v) {
    int r = row0 + v + 8 * hi;
    long ab = (long)(xidx ? xidx[r] : r) * K;
    float s = 0.f;
    for (int k = 0; k < K; ++k) s += A[ab + k] * B[k * FEAT + col0 + l16];
    c[v] = s;
  }
#endif
  float ca = coladd ? coladd[col0 + l16] : 0.0f;
#pragma unroll
  for (int v = 0; v < 8; ++v) {
    int r = row0 + v + 8 * hi;           // C/D layout: VGPR v -> row v (+8 for hi half)
    D[(long)r * FEAT + col0 + l16] = c[v] + ca;
  }
}

// ---------------- degree count & dinv = rsqrt(deg + 1) ---------------------
__global__ void deg_count_k(const int* __restrict__ dst, int E, float* __restrict__ deg) {
  int e = blockIdx.x * blockDim.x + threadIdx.x;
  if (e < E) atomicAdd(&deg[dst[e]], 1.0f);
}

__global__ void deg_to_dinv_k(float* __restrict__ deg, int n) {
  int i = blockIdx.x * blockDim.x + threadIdx.x;
  if (i < n) deg[i] = rsqrtf(deg[i] + 1.0f);
}

// ---------------- message scatter: agg[dst] += h[src] * dinv[src]*dinv[dst]
// 8 threads per edge, 4 features each.
__global__ void edge_scatter_k(const int* __restrict__ src, const int* __restrict__ dst,
                               const float* __restrict__ dinv, const float* __restrict__ h,
                               float* __restrict__ agg, int E) {
  long g = (long)blockIdx.x * blockDim.x + threadIdx.x;
  if (g >= (long)E * 8) return;
  int e = (int)(g >> 3);
  int j0 = ((int)g & 7) * 4;
  int s = src[e], d = dst[e];
  float w = dinv[s] * dinv[d];
  const float* hs = h + (long)s * FEAT + j0;
  float* ad = agg + (long)d * FEAT + j0;
#pragma unroll
  for (int j = 0; j < 4; ++j) atomicAdd(&ad[j], hs[j] * w);
}

// ---------------- agg += h * dinv^2 + bias ---------------------------------
__global__ void conv_finalize_k(const float* __restrict__ h, const float* __restrict__ dinv,
                                const float* __restrict__ bias, float* __restrict__ agg,
                                int n) {
  long g = (long)blockIdx.x * blockDim.x + threadIdx.x;
  if (g >= (long)n * FEAT) return;
  int r = (int)(g >> 5), j = (int)(g & 31);
  float di = dinv[r];
  agg[g] += h[g] * di * di + bias[j];
}

// ---------------- out = relu(x*scale + shift), store or accumulate ---------
__global__ void gn_apply_relu_k(const float* __restrict__ X, const float* __restrict__ scale,
                                const float* __restrict__ shift, float* __restrict__ out,
                                long total, int accumulate) {
  long g = (long)blockIdx.x * blockDim.x + threadIdx.x;
  if (g >= total) return;
  int c = (int)(g & 31);
  float v = fmaxf(X[g] * scale[c] + shift[c], 0.0f);
  if (accumulate) out[g] += v; else out[g] = v;
}

// ---------------- hp[p] = hr[pos[p,0]] * hr[pos[p,1]] ----------------------
__global__ void pair_prod_k(const float* __restrict__ hr, const int* __restrict__ pos,
                            float* __restrict__ hp, int P) {
  long g = (long)blockIdx.x * blockDim.x + threadIdx.x;
  if (g >= (long)P * FEAT) return;
  int p = (int)(g >> 5), j = (int)(g & 31);
  int i0 = pos[2 * p], i1 = pos[2 * p + 1];
  hp[g] = hr[(long)i0 * FEAT + j] * hr[(long)i1 * FEAT + j];
}

// ---------------- out[i] = dot(hsum[idx[2i]]*hsum[idx[2i+1]], Wp) + bp -----
__global__ void final_out_k(const float* __restrict__ hsum, const int* __restrict__ idx,
                            const float* __restrict__ Wp, const float* __restrict__ bp,
                            float* __restrict__ out, int M) {
  int i = blockIdx.x * blockDim.x + threadIdx.x;
  if (i >= M) return;
  const float* a = hsum + (long)idx[2 * i] * FEAT;
  const float* b = hsum + (long)idx[2 * i + 1] * FEAT;
  float s = 0.f;
#pragma unroll
  for (int j = 0; j < FEAT; ++j) s += a[j] * b[j] * Wp[j];
  out[i] = s + bp[0];
}

// ===========================================================================
extern "C" void kernel_launch(void* const* d_in, const int* in_sizes, int n_in,
                              void* d_out, int out_size, void* d_ws, size_t ws_size,
                              hipStream_t stream) {
  (void)in_sizes; (void)n_in; (void)out_size; (void)ws_size;
  const int N = 100000, E1 = 3200000, P = 262144, E2 = 2000000, Q = 131072;

  const int*   x     = (const int*)d_in[0];
  const int*   edge1 = (const int*)d_in[1];   // [2, E1]: src row 0, dst row 1
  const int*   pos   = (const int*)d_in[2];   // [P, 2]
  const int*   idx   = (const int*)d_in[3];   // [Q]
  const int*   ei2   = (const int*)d_in[4];   // [2, E2]
  const float* emb   = (const float*)d_in[5]; // [N, 256]
  const float* gn0_w = (const float*)d_in[6];
  const float* gn0_b = (const float*)d_in[7];
  const float* gn0_a = (const float*)d_in[8];
  const float* W1    = (const float*)d_in[9];  // [256, 32]
  const float* b1    = (const float*)d_in[10];
  const float* gn1_w = (const float*)d_in[11];
  const float* gn1_b = (const float*)d_in[12];
  const float* gn1_a = (const float*)d_in[13];
  const float* W2    = (const float*)d_in[14]; // [32, 32]
  const float* b2    = (const float*)d_in[15];
  const float* gn2_w = (const float*)d_in[16];
  const float* gn2_b = (const float*)d_in[17];
  const float* gn2_a = (const float*)d_in[18];
  const float* W2r   = (const float*)d_in[19];
  const float* b2r   = (const float*)d_in[20];
  const float* gn2r_w= (const float*)d_in[21];
  const float* gn2r_b= (const float*)d_in[22];
  const float* gn2r_a= (const float*)d_in[23];
  const float* Wp    = (const float*)d_in[24]; // [32, 1]
  const float* bp    = (const float*)d_in[25];
  float* out = (float*)d_out;

  // ---- workspace layout (floats) ----
  float* ws = (float*)d_ws;
  size_t off = 0;
  float* h1    = ws + off; off += (size_t)N * FEAT;   // gemm1 out, reused as hr
  float* agg1  = ws + off; off += (size_t)N * FEAT;
  float* deg1  = ws + off; off += (size_t)N;
  float* hp    = ws + off; off += (size_t)P * FEAT;
  float* g2    = ws + off; off += (size_t)P * FEAT;
  float* agg2  = ws + off; off += (size_t)P * FEAT;
  float* hsum  = ws + off; off += (size_t)P * FEAT;
  float* deg2  = ws + off; off += (size_t)P;
  float* stats = ws + off; off += 512;
  float* scl   = ws + off; off += 256;
  float* shf   = ws + off; off += 256;
  float* W1p   = ws + off; off += 256 * FEAT;
  float* tb    = ws + off; off += FEAT;

  auto nb = [](long n, int bs) { return (unsigned)((n + bs - 1) / bs); };

  // ---- layer 0: graph_norm(emb[x]) folded into W1 ----
  zero_f32_k<<<nb(512, 256), 256, 0, stream>>>(stats, 512);
  colstats_k<<<256, 256, 0, stream>>>(emb, x, N, 256, stats);
  gn_params_k<<<1, 256, 0, stream>>>(stats, gn0_a, gn0_w, gn0_b, 256, (float)N, scl, shf);
  prep_w1_k<<<1, 256, 0, stream>>>(W1, scl, shf, W1p, tb);

  // ---- h1 = gn0(emb[x]) @ W1  (via folded W1p, tb) ----
  {
    long waves = (long)(N / 16) * 2;
    gemm_wmma_k<<<nb(waves * 32, 256), 256, 0, stream>>>(emb, W1p, h1, tb, x, N, 256);
  }

  // ---- gcn_conv 1 over edge1 ----
  zero_f32_k<<<nb((long)N * FEAT, 256), 256, 0, stream>>>(agg1, (long)N * FEAT);
  zero_f32_k<<<nb(N, 256), 256, 0, stream>>>(deg1, N);
  deg_count_k<<<nb(E1, 256), 256, 0, stream>>>(edge1 + E1, E1, deg1);
  deg_to_dinv_k<<<nb(N, 256), 256, 0, stream>>>(deg1, N);
  edge_scatter_k<<<nb((long)E1 * 8, 256), 256, 0, stream>>>(edge1, edge1 + E1, deg1, h1, agg1, E1);
  conv_finalize_k<<<nb((long)N * FEAT, 256), 256, 0, stream>>>(h1, deg1, b1, agg1, N);

  // ---- relu(graph_norm_1) -> h1 (buffer reuse) ----
  zero_f32_k<<<nb(512, 256), 256, 0, stream>>>(stats, 512);
  colstats_k<<<256, 256, 0, stream>>>(agg1, nullptr, N, 32, stats);
  gn_params_k<<<1, 32, 0, stream>>>(stats, gn1_a, gn1_w, gn1_b, 32, (float)N, scl, shf);
  gn_apply_relu_k<<<nb((long)N * FEAT, 256), 256, 0, stream>>>(agg1, scl, shf, h1,
                                                               (long)N * FEAT, 0);

  // ---- hp = h1[pos0] * h1[pos1] ----
  pair_prod_k<<<nb((long)P * FEAT, 256), 256, 0, stream>>>(h1, pos, hp, P);

  // ---- two gcn branches on ei2 (forward / reversed) ----
  auto branch = [&](const int* src, const int* dst, const float* W, const float* bb,
                    const float* ga, const float* gw, const float* gb, int accum) {
    long waves = (long)(P / 16) * 2;
    gemm_wmma_k<<<nb(waves * 32, 256), 256, 0, stream>>>(hp, W, g2, nullptr, nullptr, P, 32);
    zero_f32_k<<<nb((long)P * FEAT, 256), 256, 0, stream>>>(agg2, (long)P * FEAT);
    zero_f32_k<<<nb(P, 256), 256, 0, stream>>>(deg2, P);
    deg_count_k<<<nb(E2, 256), 256, 0, stream>>>(dst, E2, deg2);
    deg_to_dinv_k<<<nb(P, 256), 256, 0, stream>>>(deg2, P);
    edge_scatter_k<<<nb((long)E2 * 8, 256), 256, 0, stream>>>(src, dst, deg2, g2, agg2, E2);
    conv_finalize_k<<<nb((long)P * FEAT, 256), 256, 0, stream>>>(g2, deg2, bb, agg2, P);
    zero_f32_k<<<nb(512, 256), 256, 0, stream>>>(stats, 512);
    colstats_k<<<256, 256, 0, stream>>>(agg2, nullptr, P, 32, stats);
    gn_params_k<<<1, 32, 0, stream>>>(stats, ga, gw, gb, 32, (float)P, scl, shf);
    gn_apply_relu_k<<<nb((long)P * FEAT, 256), 256, 0, stream>>>(agg2, scl, shf, hsum,
                                                                 (long)P * FEAT, accum);
  };
  branch(ei2,      ei2 + E2, W2,  b2,  gn2_a,  gn2_w,  gn2_b,  0);
  branch(ei2 + E2, ei2,      W2r, b2r, gn2r_a, gn2r_w, gn2r_b, 1);

  // ---- gather idx, pairwise product, @ Wp + bp ----
  final_out_k<<<nb(Q / 2, 256), 256, 0, stream>>>(hsum, idx, Wp, bp, out, Q / 2);
}